// HFsolver_16097537425872
// MI455X (gfx1250) — compile-verified
//
#include <hip/hip_runtime.h>

#define NXG       240
#define NTHREADS  512     // 16 waves of 32; waves 0..14 own the 15 M-tiles of 16 rows
#define NKT       60      // 240 / 4 K-steps per WMMA product
#define FFACC     0.4f

typedef __attribute__((ext_vector_type(2))) float v2f;
typedef __attribute__((ext_vector_type(8))) float v8f;

// Block-wide sum reduction with broadcast: wave32 shuffle partials + 16-entry LDS combine.
__device__ __forceinline__ float block_reduce(float v, float* red, int tid) {
    #pragma unroll
    for (int off = 16; off > 0; off >>= 1)
        v += __shfl_down(v, off, 32);
    __syncthreads();                       // protect red[] reuse from previous call
    if ((tid & 31) == 0) red[tid >> 5] = v;
    __syncthreads();
    if (tid < 16) {
        float p = red[tid];
        #pragma unroll
        for (int off = 8; off > 0; off >>= 1)
            p += __shfl_down(p, off, 32);
        if (tid == 0) red[0] = p;
    }
    __syncthreads();
    return red[0];                         // next call's first barrier protects this read
}

// Y(240x2) = A(240x240) @ W(240x16, cols 0/1 = wA,wB, rest zero), via
// V_WMMA_F32_16X16X4_F32.
//  FUSED_H=true : A = kin - delx*Vint*(wA0 wA0' + wB0 wB0') ; the diagonal
//                 (Udir+U_HO) is applied at writeback as diag[i]*W[i][n].
//  FUSED_H=false: A = kin.
// sW is a zero-padded 16x240 column-major W tile in LDS (only cols 0/1 written).
// Layouts per CDNA5 ISA 7.12.2:
//   A 16x4 : lanes 0-15 hold K=0(v0),1(v1); lanes 16-31 hold K=2(v0),3(v1); M=lane&15
//   B 4x16 : same half-lane K split; N = lane&15
//   C 16x16: VGPR r -> rows M=r (lanes 0-15) and M=r+8 (lanes 16-31), N = lane&15
template <bool FUSED_H>
__device__ __forceinline__ void matprod(const float* __restrict__ kin,
                                        const float* __restrict__ Vint,
                                        float delx,
                                        const float* sA, const float* sB,
                                        const float* sDiag, float* sW,
                                        const float* wA, const float* wB,
                                        float* yA, float* yB, int tid) {
    if (tid < NXG) { sW[tid] = wA[tid]; sW[NXG + tid] = wB[tid]; }
    __syncthreads();                       // sW + all LDS operands complete
    const int wave = tid >> 5;
    const int lane = tid & 31;
    if (wave < 15) {
        const int half = lane >> 4;        // 0: K=0,1   1: K=2,3
        const int l16  = lane & 15;
        const int mrow = wave * 16 + l16;
        const float ra = sA[mrow], rb = sB[mrow];
        const float* __restrict__ krow = kin  + mrow * NXG;
        const float* __restrict__ vrow = Vint + mrow * NXG;
        const float* __restrict__ bcol = sW + l16 * NXG;
        v8f c = {};
        #pragma unroll 4
        for (int kt = 0; kt < NKT; ++kt) {
            const int k0 = kt * 4 + 2 * half;
            v2f a;
            if (FUSED_H) {
                a[0] = krow[k0]     - delx * vrow[k0]     * (ra * sA[k0]     + rb * sB[k0]);
                a[1] = krow[k0 + 1] - delx * vrow[k0 + 1] * (ra * sA[k0 + 1] + rb * sB[k0 + 1]);
            } else {
                a[0] = krow[k0];
                a[1] = krow[k0 + 1];
            }
            v2f b; b[0] = bcol[k0]; b[1] = bcol[k0 + 1];   // contiguous ds_load_b64
            c = __builtin_amdgcn_wmma_f32_16x16x4_f32(
                    /*neg_a=*/false, a, /*neg_b=*/false, b,
                    /*c_mod=*/(short)0, c, /*reuse_a=*/false, /*reuse_b=*/false);
        }
        if (l16 < 2) {                     // only columns 0/1 are meaningful
            float* dst = (l16 == 0) ? yA : yB;
            const float* wcol = sW + l16 * NXG;
            #pragma unroll
            for (int r = 0; r < 8; ++r) {
                const int row = wave * 16 + r + 8 * half;
                const float dadd = FUSED_H ? sDiag[row] * wcol[row] : 0.0f;
                dst[row] = c[r] + dadd;
            }
        }
    }
    __syncthreads();
}

__global__ void __launch_bounds__(NTHREADS)
hf_kernel(const float* __restrict__ wfy0, const float* __restrict__ kin,
          const float* __restrict__ Vint, const float* __restrict__ U_HO,
          const float* __restrict__ delx_p, const float* __restrict__ pfac_p,
          const int* __restrict__ iter_p, float* __restrict__ out) {
    __shared__ float sA[NXG], sB[NXG];      // current orbitals
    __shared__ float sDiag[NXG];            // Udir + U_HO
    __shared__ float sHden[NXG];            // density
    __shared__ float sYA[NXG], sYB[NXG];    // product output (H@x / Hn)
    __shared__ float sNA[NXG], sNB[NXG];    // mixed new orbitals
    __shared__ float sKA[NXG], sKB[NXG];    // kin@new
    __shared__ float sW[16 * NXG];          // zero-padded column-major B operand
    __shared__ float red[16];

    const int tid = threadIdx.x;
    const float delx = delx_p[0];
    const float pfac = pfac_p[0];
    const int itermax = iter_p[0];

    for (int i = tid; i < 16 * NXG; i += NTHREADS) sW[i] = 0.0f;
    if (tid < NXG) { sA[tid] = wfy0[tid * 2 + 0]; sB[tid] = wfy0[tid * 2 + 1]; }
    __syncthreads();

    float enerhf = 0.f, enerhfp = 0.f, ekin = 0.f, eho = 0.f,
          epot0hf = 0.f, esum = 0.f;

    for (int it = 0; it < itermax; ++it) {
        // density
        if (tid < NXG) sHden[tid] = sA[tid] * sA[tid] + sB[tid] * sB[tid];
        __syncthreads();
        // Udir + U_HO (diagonal of Umf)
        if (tid < NXG) {
            float acc = 0.f;
            const float* vr = Vint + tid * NXG;
            for (int j = 0; j < NXG; ++j) acc += vr[j] * sHden[j];
            sDiag[tid] = delx * acc + U_HO[tid];
        }
        // eho = sum(hf_den * U_HO) * delx / 2   (barriers also publish sDiag)
        float ev = (tid < NXG) ? sHden[tid] * U_HO[tid] : 0.f;
        eho = block_reduce(ev, red, tid) * delx * 0.5f;

        // wff = H @ wfy (both orbitals at once)
        matprod<true>(kin, Vint, delx, sA, sB, sDiag, sW, sA, sB, sYA, sYB, tid);

        // single-particle energies (pre-damping)
        float s0 = (tid < NXG) ? sA[tid] * sYA[tid] : 0.f;
        float spe0 = block_reduce(s0, red, tid) * delx;
        float s1 = (tid < NXG) ? sB[tid] * sYB[tid] : 0.f;
        float spe1 = block_reduce(s1, red, tid) * delx;

        // orbital 0: damp, normalize (sqrt), renormalize (reference divides by norm!)
        float w0 = (tid < NXG) ? (sA[tid] - pfac * sYA[tid]) : 0.f;
        float n0 = block_reduce(w0 * w0, red, tid) * delx;
        w0 /= sqrtf(n0);
        float n0b = block_reduce(w0 * w0, red, tid) * delx;
        w0 /= n0b;
        if (tid < NXG) sNA[tid] = FFACC * w0 + (1.f - FFACC) * sA[tid];

        // orbital 1: damp, normalize, Gram-Schmidt vs mixed new0, renormalize
        float w1 = (tid < NXG) ? (sB[tid] - pfac * sYB[tid]) : 0.f;
        float n1 = block_reduce(w1 * w1, red, tid) * delx;
        w1 /= sqrtf(n1);
        float na = (tid < NXG) ? sNA[tid] : 0.f;
        float dot = block_reduce(na * w1, red, tid) * delx;
        w1 -= na * dot;
        float n1b = block_reduce(w1 * w1, red, tid) * delx;
        w1 /= n1b;
        if (tid < NXG) sNB[tid] = FFACC * w1 + (1.f - FFACC) * sB[tid];

        // kin@new and H@new  (Umf@new = H@new - kin@new; H still uses OLD orbitals)
        matprod<false>(kin, Vint, delx, sA, sB, sDiag, sW, sNA, sNB, sKA, sKB, tid);
        matprod<true >(kin, Vint, delx, sA, sB, sDiag, sW, sNA, sNB, sYA, sYB, tid);

        float kv = (tid < NXG) ? (sNA[tid] * sKA[tid] + sNB[tid] * sKB[tid]) : 0.f;
        ekin = block_reduce(kv, red, tid) * delx;
        float pv = (tid < NXG) ? (sNA[tid] * (sYA[tid] - sKA[tid]) +
                                  sNB[tid] * (sYB[tid] - sKB[tid])) : 0.f;
        float epot = block_reduce(pv, red, tid) * delx;

        esum    = spe0 + spe1;
        enerhfp = (esum + ekin) * 0.5f + eho;
        epot0hf = epot - 2.f * eho;
        enerhf  = esum - epot0hf * 0.5f;

        __syncthreads();
        if (tid < NXG) { sA[tid] = sNA[tid]; sB[tid] = sNB[tid]; }
        __syncthreads();
    }

    if (tid == 0) {
        out[0] = enerhf; out[1] = enerhfp; out[2] = ekin;
        out[3] = eho;    out[4] = epot0hf; out[5] = esum;
    }
}

extern "C" void kernel_launch(void* const* d_in, const int* in_sizes, int n_in,
                              void* d_out, int out_size, void* d_ws, size_t ws_size,
                              hipStream_t stream) {
    (void)in_sizes; (void)n_in; (void)out_size; (void)d_ws; (void)ws_size;
    const float* wfy0 = (const float*)d_in[0];
    const float* kin  = (const float*)d_in[1];
    const float* Vint = (const float*)d_in[2];
    const float* U_HO = (const float*)d_in[3];
    const float* delx = (const float*)d_in[4];
    const float* pfac = (const float*)d_in[5];
    const int*   itm  = (const int*)d_in[6];
    hf_kernel<<<1, NTHREADS, 0, stream>>>(wfy0, kin, Vint, U_HO, delx, pfac, itm,
                                          (float*)d_out);
}